// Downsample_7876970021203
// MI455X (gfx1250) — compile-verified
//
#include <hip/hip_runtime.h>

typedef __attribute__((ext_vector_type(2))) float v2f;
typedef __attribute__((ext_vector_type(8))) float v8f;

// Problem constants (match setup_inputs in the reference)
#define B_ 8
#define N_ 8192
#define C_ 64
#define M_ 2048   // N_/stride, stride=4
#define K_ 16     // num_neighbors

// ---------------------------------------------------------------------------
// Kernel 1: Furthest Point Sampling. One workgroup per batch.
// Coordinates live in registers (8 points/thread); an LDS copy provides
// random-index broadcast of the last selected point. Also writes p1 SoA +
// squared norms into workspace for the WMMA KNN kernel, and writes p2/idx
// outputs directly.
// ---------------------------------------------------------------------------
__global__ __launch_bounds__(1024) void fps_kernel(
    const float* __restrict__ p1,
    float* __restrict__ out_p2,   // (B,M,3)
    int*   __restrict__ out_idx,  // (B,M)
    float* __restrict__ wsx, float* __restrict__ wsy,
    float* __restrict__ wsz, float* __restrict__ wsn) {
  extern __shared__ float lds[];                 // 3*N_ floats (96 KB of 320 KB/WGP)
  float* sx = lds;
  float* sy = lds + N_;
  float* sz = lds + 2 * N_;
  __shared__ float redv[32];
  __shared__ int   redi[32];
  __shared__ int   s_last;

  const int b   = blockIdx.x;
  const int tid = threadIdx.x;
  const float* pb = p1 + (size_t)b * N_ * 3;

  float px[8], py[8], pz[8], d2[8];
#pragma unroll
  for (int s = 0; s < 8; ++s) {
    int pid = tid + s * 1024;
    float x = pb[pid * 3 + 0];
    float y = pb[pid * 3 + 1];
    float z = pb[pid * 3 + 2];
    px[s] = x; py[s] = y; pz[s] = z;
    d2[s] = __builtin_inff();
    sx[pid] = x; sy[pid] = y; sz[pid] = z;
    size_t g = (size_t)b * N_ + pid;
    wsx[g] = x; wsy[g] = y; wsz[g] = z;
    wsn[g] = x * x + y * y + z * z;              // |p1|^2 for KNN scores
  }
  __syncthreads();

  const int lane = tid & 31;
  const int wid  = tid >> 5;
  int last = 0;
  for (int m = 0; m < M_; ++m) {
    float lx = sx[last], ly = sy[last], lz = sz[last];   // uniform LDS broadcast
    if (tid == 0) {
      size_t o = (size_t)b * M_ + m;
      out_idx[o] = last;
      out_p2[o * 3 + 0] = lx;
      out_p2[o * 3 + 1] = ly;
      out_p2[o * 3 + 2] = lz;
    }
    float bv = -1.0f; int bi = 0;
#pragma unroll
    for (int s = 0; s < 8; ++s) {
      float dx = px[s] - lx, dy = py[s] - ly, dz = pz[s] - lz;
      float d = dx * dx + dy * dy + dz * dz;
      d = fminf(d2[s], d);
      d2[s] = d;
      int pid = tid + s * 1024;
      if (d > bv || (d == bv && pid < bi)) { bv = d; bi = pid; }
    }
    // wave32 argmax (ties -> lowest index, matching jnp.argmax)
#pragma unroll
    for (int off = 16; off >= 1; off >>= 1) {
      float ov = __shfl_xor(bv, off, 32);
      int   oi = __shfl_xor(bi, off, 32);
      if (ov > bv || (ov == bv && oi < bi)) { bv = ov; bi = oi; }
    }
    if (lane == 0) { redv[wid] = bv; redi[wid] = bi; }
    __syncthreads();
    if (wid == 0) {
      bv = redv[lane]; bi = redi[lane];
#pragma unroll
      for (int off = 16; off >= 1; off >>= 1) {
        float ov = __shfl_xor(bv, off, 32);
        int   oi = __shfl_xor(bi, off, 32);
        if (ov > bv || (ov == bv && oi < bi)) { bv = ov; bi = oi; }
      }
      if (lane == 0) s_last = bi;
    }
    __syncthreads();
    last = s_last;
  }
}

// ---------------------------------------------------------------------------
// Kernel 2: KNN via V_WMMA_F32_16X16X4_F32 distance tiles + distributed
// top-16 selection + fused gather-mean.
// One wave owns 16 queries; 4 waves per block; grid = B*M/64 blocks.
// Score used for ranking: |p1|^2 - 2*p2.p1   (|p2|^2 constant per row).
// ---------------------------------------------------------------------------
__global__ __launch_bounds__(128) void knn_mean_kernel(
    const float* __restrict__ x,    // (B,N,C)
    const float* __restrict__ p2,   // (B,M,3) written by fps_kernel
    const float* __restrict__ wsx, const float* __restrict__ wsy,
    const float* __restrict__ wsz, const float* __restrict__ wsn,
    float* __restrict__ out_y) {    // (B,M,C)
  __shared__ int s_nn[4][16 * K_];

  const int lane = threadIdx.x & 31;
  const int wv   = threadIdx.x >> 5;
  const int gw   = blockIdx.x * 4 + wv;
  const int b    = gw >> 7;            // M_/16 = 128 query tiles per batch
  const int tile = gw & 127;
  const int q0   = tile * 16;
  const int qm   = lane & 15;
  const bool hi  = lane >= 16;

  // A matrix (16 queries x 4): lanes 0-15 hold K0=x,K1=y; lanes 16-31 K2=z,K3=0
  size_t p2o = ((size_t)b * M_ + (q0 + qm)) * 3;
  v2f A;
  A.x = hi ? p2[p2o + 2] : p2[p2o + 0];
  A.y = hi ? 0.0f        : p2[p2o + 1];

  // Distributed top-16 per query row: each lane of a 16-lane half owns 1 slot.
  float kmax[8], slotval[8];
  int   kslot[8], slotidx[8];
#pragma unroll
  for (int r = 0; r < 8; ++r) {
    kmax[r] = __builtin_inff(); kslot[r] = 0;
    slotval[r] = __builtin_inff(); slotidx[r] = 0;
  }

  const size_t bn = (size_t)b * N_;
  for (int n0 = 0; n0 < N_; n0 += 16) {
    const int col = n0 + qm;
    // B matrix (4 x 16 points), layout mirrored from the documented A layout
    v2f Bt;
    Bt.x = hi ? wsz[bn + col] : wsx[bn + col];
    Bt.y = hi ? 0.0f          : wsy[bn + col];
    float nrm = wsn[bn + col];

    v8f cz = {0.f, 0.f, 0.f, 0.f, 0.f, 0.f, 0.f, 0.f};
    // v_wmma_f32_16x16x4_f32: acc[r] @ lane L = dot(p2[r+8*(L>=16)], p1[n0+(L&15)])
    v8f acc = __builtin_amdgcn_wmma_f32_16x16x4_f32(
        false, A, false, Bt, (short)0, cz, false, false);

#pragma unroll
    for (int r = 0; r < 8; ++r) {
      float sc = __builtin_fmaf(-2.0f, acc[r], nrm);
      unsigned long long qual = __ballot(sc < kmax[r]);
      while (qual) {                      // uniform loop: EXEC stays all-ones
        int bl = __ffsll((long long)qual) - 1;
        qual &= qual - 1;
        float val = __shfl(sc, bl, 32);
        int  vidx = __shfl(col, bl, 32);  // col of lane bl
        bool inHalf = ((lane ^ bl) & 16) == 0;
        if (inHalf && val < kmax[r]) {
          if ((lane & 15) == kslot[r]) { slotval[r] = val; slotidx[r] = vidx; }
        }
        // recompute kth-max + its slot within each 16-lane half (xor<=8 stays in half)
        float v = slotval[r]; int li = lane & 15;
#pragma unroll
        for (int d = 8; d >= 1; d >>= 1) {
          float ov = __shfl_xor(v, d, 32);
          int  oli = __shfl_xor(li, d, 32);
          if (ov > v || (ov == v && oli < li)) { v = ov; li = oli; }
        }
        kmax[r] = v; kslot[r] = li;
      }
    }
  }

  // Publish neighbor sets: query row Ml = r + 8*hi, slot = lane&15
#pragma unroll
  for (int r = 0; r < 8; ++r) {
    int Ml = r + (hi ? 8 : 0);
    s_nn[wv][Ml * K_ + (lane & 15)] = slotidx[r];
  }
  __syncthreads();

  // Fused gather + mean over k=16 (order-invariant, so unsorted set is fine).
  // Lane c handles channels c and c+32 -> coalesced feature reads.
  for (int qq = 0; qq < 16; ++qq) {
    float a0 = 0.f, a1 = 0.f;
#pragma unroll
    for (int j = 0; j < K_; ++j) {
      int nn = s_nn[wv][qq * K_ + j];
      const float* xr = x + ((size_t)b * N_ + nn) * C_;
      a0 += xr[lane];
      a1 += xr[lane + 32];
    }
    size_t yo = ((size_t)b * M_ + (q0 + qq)) * C_;
    out_y[yo + lane]      = a0 * (1.0f / 16.0f);
    out_y[yo + lane + 32] = a1 * (1.0f / 16.0f);
  }
}

// ---------------------------------------------------------------------------
extern "C" void kernel_launch(void* const* d_in, const int* in_sizes, int n_in,
                              void* d_out, int out_size, void* d_ws, size_t ws_size,
                              hipStream_t stream) {
  const float* p1 = (const float*)d_in[0];
  const float* x  = (const float*)d_in[1];
  // d_in[2]=stride(4), d_in[3]=num_neighbors(16): fixed by setup_inputs.

  float* out   = (float*)d_out;
  float* y     = out;                                   // (B,M,C)
  float* p2    = out + (size_t)B_ * M_ * C_;            // (B,M,3)
  int*   idx   = (int*)(p2 + (size_t)B_ * M_ * 3);      // (B,M) int32

  float* wsx = (float*)d_ws;                            // p1 SoA + norms: 1 MB
  float* wsy = wsx + (size_t)B_ * N_;
  float* wsz = wsy + (size_t)B_ * N_;
  float* wsn = wsz + (size_t)B_ * N_;

  fps_kernel<<<B_, 1024, 3 * N_ * sizeof(float), stream>>>(
      p1, p2, idx, wsx, wsy, wsz, wsn);

  const int waves  = B_ * (M_ / 16);   // 1024 query tiles
  knn_mean_kernel<<<waves / 4, 128, 0, stream>>>(
      x, p2, wsx, wsy, wsz, wsn, y);
}